// Model_83038897701197
// MI455X (gfx1250) — compile-verified
//
#include <hip/hip_runtime.h>
#include <math.h>

typedef __attribute__((ext_vector_type(2))) float v2f;
typedef __attribute__((ext_vector_type(4))) float v4f;
typedef __attribute__((ext_vector_type(8))) float v8f;

#define B_    32
#define HKV   8
#define GQ    4
#define D_    128
#define PS    16
#define NW    8            // waves per workgroup (wave32)
#define KROW  132          // padded floats per K row in LDS (132*4 = 528 B, 16B aligned)
#define KTILE (PS * KROW)  // 2112 floats per wave staging region (>= 512 needed for merge)
#define NEG_BIG (-1000000.0f)

static __device__ inline v2f mk2(float x, float y) { v2f r; r.x = x; r.y = y; return r; }

__global__ __launch_bounds__(NW * 32) void paged_attn_decode(
    const float* __restrict__ q,        // [B][HKV*GQ][1][D]
    const float* __restrict__ kv,       // [page][2][HKV][PS][D]
    const int*   __restrict__ indptr,   // [B+1]
    const int*   __restrict__ pidx,     // [B*PAGES]
    const int*   __restrict__ lastlen,  // [B]
    float*       __restrict__ out)      // [B][HKV*GQ][1][D]
{
    extern __shared__ float lds[];      // NW * KTILE floats (K staging, reused for merge)
    __shared__ float lds_m[NW][GQ];
    __shared__ float lds_l[NW][GQ];

    const int tid  = threadIdx.x;
    const int lane = tid & 31;
    const int w    = tid >> 5;
    const int m16  = lane & 15;
    const int hi   = lane >> 4;

    const int b   = blockIdx.x >> 3;    // HKV = 8
    const int kvh = blockIdx.x & 7;

    const int p0   = indptr[b];
    const int np   = indptr[b + 1] - p0;
    const int llen = lastlen[b];

    // ---- Q in B-operand layout (Q^T: k = dim chunk, n = head), softmax scale folded in.
    const float scale = 0.08838834764831845f;   // 128^-0.5
    v2f qa[32];
    {
        const float* qp = q + ((size_t)b * (HKV * GQ) + (size_t)kvh * GQ + m16) * D_;
        #pragma unroll
        for (int kc = 0; kc < 32; ++kc) {
            if (m16 < GQ) {
                const int dbase = kc * 4 + 2 * hi;
                qa[kc] = mk2(qp[dbase] * scale, qp[dbase + 1] * scale);
            } else {
                qa[kc] = mk2(0.f, 0.f);     // zero garbage head rows -> finite garbage outputs
            }
        }
    }

    v8f acc[8] = {};                     // out^T tiles: elem r of lane -> (dim dc*16+r+8*hi, head m16)
    float m_run = -1e30f;
    float l_run = 0.f;

    float* myK = lds + (size_t)w * KTILE;

    for (int i = w; i < np; i += NW) {
        const int page = pidx[p0 + i];
        const float* Kb = kv + (size_t)page * (2 * HKV * PS * D_) + (size_t)kvh * (PS * D_);
        const float* Vb = Kb + (size_t)(HKV * PS * D_);

        // ---- prefetch this wave's next page (4 KB per prefetch, 128B lane stride)
        if (i + NW < np) {
            const int pn = pidx[p0 + i + NW];
            const float* nK = kv + (size_t)pn * (2 * HKV * PS * D_) + (size_t)kvh * (PS * D_);
            __builtin_prefetch(nK + lane * 32, 0, 0);
            __builtin_prefetch(nK + 1024 + lane * 32, 0, 0);
            __builtin_prefetch(nK + (HKV * PS * D_) + lane * 32, 0, 0);
            __builtin_prefetch(nK + (HKV * PS * D_) + 1024 + lane * 32, 0, 0);
        }

        // ---- stage K page -> LDS, coalesced 512B non-temporal b128 loads, padded rows
        #pragma unroll
        for (int t = 0; t < PS; ++t) {
            v4f kval = __builtin_nontemporal_load((const v4f*)(Kb + t * D_ + lane * 4));
            *(v4f*)(myK + t * KROW + lane * 4) = kval;
        }

        // ---- S^T = K_tile x Q^T  (M = token, N = head), 32 k-steps of 4 dims
        v8f cs = {};
        const float* krow = myK + m16 * KROW + 2 * hi;   // A: K[token = m16][kc*4 + 2*hi + j]
        #pragma unroll
        for (int kc = 0; kc < 32; ++kc) {
            v2f a = *(const v2f*)(krow + kc * 4);
            cs = __builtin_amdgcn_wmma_f32_16x16x4_f32(false, a, false, qa[kc],
                                                       (short)0, cs, false, false);
        }

        // ---- mask last page, scores: elem r -> token r + 8*hi, head m16
        float s[8];
        #pragma unroll
        for (int r = 0; r < 8; ++r) s[r] = cs[r];
        if (i == np - 1) {
            #pragma unroll
            for (int r = 0; r < 8; ++r)
                if (r + 8 * hi >= llen) s[r] += NEG_BIG;
        }

        // ---- online softmax: per-lane head state, one shfl to merge token halves
        float pm = s[0];
        #pragma unroll
        for (int r = 1; r < 8; ++r) pm = fmaxf(pm, s[r]);
        pm = fmaxf(pm, __shfl_xor(pm, 16, 32));
        const float m_new = fmaxf(m_run, pm);
        const float alpha = __expf(m_run - m_new);
        m_run = m_new;

        float p[8], psum = 0.f;
        #pragma unroll
        for (int r = 0; r < 8; ++r) { p[r] = __expf(s[r] - m_new); psum += p[r]; }
        l_run = l_run * alpha + psum;

        #pragma unroll
        for (int dc = 0; dc < 8; ++dc)
            #pragma unroll
            for (int r = 0; r < 8; ++r) acc[dc][r] *= alpha;

        // ---- P^T B-operands: lane needs p[head=m16][tok = tk*4 + 2*hi + j]; half-swap via shfl
        float psw[8];
        #pragma unroll
        for (int r = 0; r < 8; ++r) psw[r] = __shfl_xor(p[r], 16, 32);

        v2f bp[4];
        bp[0] = hi ? mk2(psw[2], psw[3]) : mk2(p[0], p[1]);
        bp[1] = hi ? mk2(psw[6], psw[7]) : mk2(p[4], p[5]);
        bp[2] = hi ? mk2(p[2], p[3])     : mk2(psw[0], psw[1]);
        bp[3] = hi ? mk2(p[6], p[7])     : mk2(psw[4], psw[5]);

        // ---- out^T += V^T x P^T  (A: V[tok][dc*16 + m16], 64B contiguous per half-wave, NT)
        #pragma unroll
        for (int tk = 0; tk < 4; ++tk) {
            const int t0 = tk * 4 + 2 * hi;
            const float* v0 = Vb + t0 * D_ + m16;
            #pragma unroll
            for (int dc = 0; dc < 8; ++dc) {
                v2f av = mk2(__builtin_nontemporal_load(v0 + dc * 16),
                             __builtin_nontemporal_load(v0 + dc * 16 + D_));
                acc[dc] = __builtin_amdgcn_wmma_f32_16x16x4_f32(false, av, false, bp[tk],
                                                                (short)0, acc[dc], false, false);
            }
        }
    }

    // ---- publish per-wave partials (m, L, acc) into LDS; acc reuses this wave's K region
    const float l_tot = l_run + __shfl_xor(l_run, 16, 32);
    if (m16 < GQ && hi == 0) { lds_m[w][m16] = m_run; lds_l[w][m16] = l_tot; }
    if (m16 < GQ) {
        float* ab = lds + (size_t)w * KTILE + m16 * D_;
        #pragma unroll
        for (int dc = 0; dc < 8; ++dc)
            #pragma unroll
            for (int r = 0; r < 8; ++r)
                ab[dc * 16 + r + 8 * hi] = acc[dc][r];
    }
    __syncthreads();

    // ---- merge NW wave-partials (flash-decode combine) and write output
    for (int t = tid; t < GQ * D_; t += NW * 32) {
        const int h = t >> 7, d = t & 127;
        float mg = -1e30f;
        #pragma unroll
        for (int wv = 0; wv < NW; ++wv) mg = fmaxf(mg, lds_m[wv][h]);
        float Lg = 0.f, o = 0.f;
        #pragma unroll
        for (int wv = 0; wv < NW; ++wv) {
            const float f = __expf(lds_m[wv][h] - mg);
            Lg += lds_l[wv][h] * f;
            o  += lds[(size_t)wv * KTILE + h * D_ + d] * f;
        }
        out[((size_t)b * (HKV * GQ) + (size_t)kvh * GQ + h) * D_ + d] = o / Lg;
    }
}

extern "C" void kernel_launch(void* const* d_in, const int* in_sizes, int n_in,
                              void* d_out, int out_size, void* d_ws, size_t ws_size,
                              hipStream_t stream) {
    (void)in_sizes; (void)n_in; (void)out_size; (void)d_ws; (void)ws_size;
    const float* q       = (const float*)d_in[0];
    const float* kv      = (const float*)d_in[1];
    const int*   indptr  = (const int*)d_in[2];
    const int*   pidx    = (const int*)d_in[3];
    const int*   lastlen = (const int*)d_in[4];
    float*       out     = (float*)d_out;

    dim3 grid(B_ * HKV);
    dim3 block(NW * 32);
    size_t shmem = (size_t)NW * KTILE * sizeof(float);   // 67,584 B < 320 KB/WGP
    hipLaunchKernelGGL(paged_attn_decode, grid, block, shmem, stream,
                       q, kv, indptr, pidx, lastlen, out);
}